// GaussianRasterizer_58205396795780
// MI455X (gfx1250) — compile-verified
//
#include <hip/hip_runtime.h>
#include <math.h>

// ---------------------------------------------------------------------------
// Gaussian splat rasterizer for MI455X (gfx1250, wave32).
// power(n,p) = A(n) . B(p), K=6 (pad 8) -> V_WMMA_F32_16X16X4_F32 x2 chained.
// One wave = 32 pixels (two 16-pixel WMMA tiles); transmittance scan in regs.
// A operands are loaded as lane-addressed v2f slices (no select trees).
// ---------------------------------------------------------------------------

#define N_G    1024
#define W_IMG  128
#define H_IMG  128
#define FOCALF 128.0f      // W/(2*tanfovx) = H/(2*tanfovy) = 128
#define LIMXY  0.65f       // 1.3 * tanfov

typedef __attribute__((ext_vector_type(2))) float v2f;
typedef __attribute__((ext_vector_type(8))) float v8f;

// workspace layout (float offsets)
#define OFF_KEY   0
#define OFF_A     1024          // 1024*8
#define OFF_COL   9216          // 1024*3
#define OFF_OP    12288         // 1024
#define OFF_TZ    13312         // 1024
#define OFF_SA    14336         // sorted A, 1024*8 (32B aligned base)
#define OFF_SCOL  22528         // sorted color, 1024*3
#define OFF_SOP   25600         // sorted opacity (valid-gated)
#define OFF_STZ   26624         // sorted view-depth
// total 27648 floats = 108 KB

// d_out layout (floats): color 3*16384 | depth 16384 | alpha 16384 | radii 1024 | p_view 3072
#define OUT_DEPTH 49152
#define OUT_ALPHA 65536
#define OUT_RADII 81920
#define OUT_PVIEW 82944

// ---------------------------- kernel 1: preprocess -------------------------
__global__ void preprocess_kernel(const float* __restrict__ means,
                                  const float* __restrict__ cols,
                                  const float* __restrict__ ops,
                                  const float* __restrict__ scales,
                                  const float* __restrict__ rots,
                                  const float* __restrict__ vm,
                                  const float* __restrict__ pm,
                                  float* __restrict__ ws,
                                  float* __restrict__ out)
{
    int i = blockIdx.x * blockDim.x + threadIdx.x;
    if (i >= N_G) return;

    float mx = means[i*3+0], my = means[i*3+1], mz = means[i*3+2];

    // p_view = ([m,1] @ viewmatrix)[:3]
    float pv0 = mx*vm[0] + my*vm[4] + mz*vm[8]  + vm[12];
    float pv1 = mx*vm[1] + my*vm[5] + mz*vm[9]  + vm[13];
    float tz  = mx*vm[2] + my*vm[6] + mz*vm[10] + vm[14];

    // p_hom = [m,1] @ projmatrix
    float h0 = mx*pm[0] + my*pm[4] + mz*pm[8]  + pm[12];
    float h1 = mx*pm[1] + my*pm[5] + mz*pm[9]  + pm[13];
    float h3 = mx*pm[3] + my*pm[7] + mz*pm[11] + pm[15];
    float pwv = 1.0f / (h3 + 1e-7f);
    float ndx = h0 * pwv, ndy = h1 * pwv;

    // quaternion -> rotation
    float qr = rots[i*4+0], qx = rots[i*4+1], qy = rots[i*4+2], qz = rots[i*4+3];
    float qinv = 1.0f / sqrtf(qr*qr + qx*qx + qy*qy + qz*qz);
    qr *= qinv; qx *= qinv; qy *= qinv; qz *= qinv;
    float R00 = 1.f - 2.f*(qy*qy + qz*qz), R01 = 2.f*(qx*qy - qr*qz), R02 = 2.f*(qx*qz + qr*qy);
    float R10 = 2.f*(qx*qy + qr*qz), R11 = 1.f - 2.f*(qx*qx + qz*qz), R12 = 2.f*(qy*qz - qr*qx);
    float R20 = 2.f*(qx*qz - qr*qy), R21 = 2.f*(qy*qz + qr*qx), R22 = 1.f - 2.f*(qx*qx + qy*qy);

    float s0 = scales[i*3+0], s1 = scales[i*3+1], s2 = scales[i*3+2];
    float t0 = s0*s0, t1 = s1*s1, t2 = s2*s2;   // SCALE_MOD = 1

    // Sigma = R diag(s^2) R^T (symmetric)
    float S00 = R00*R00*t0 + R01*R01*t1 + R02*R02*t2;
    float S01 = R00*R10*t0 + R01*R11*t1 + R02*R12*t2;
    float S02 = R00*R20*t0 + R01*R21*t1 + R02*R22*t2;
    float S11 = R10*R10*t0 + R11*R11*t1 + R12*R12*t2;
    float S12 = R10*R20*t0 + R11*R21*t1 + R12*R22*t2;
    float S22 = R20*R20*t0 + R21*R21*t1 + R22*R22*t2;

    float txc = fminf(fmaxf(pv0 / tz, -LIMXY), LIMXY) * tz;
    float tyc = fminf(fmaxf(pv1 / tz, -LIMXY), LIMXY) * tz;
    float invz = 1.0f / tz;
    float J00 = FOCALF * invz, J02 = -FOCALF * txc * invz * invz;
    float J11 = FOCALF * invz, J12 = -FOCALF * tyc * invz * invz;

    // Tm = J @ Wrot,  Wrot[k][j] = vm[j*4+k]
    float Tm00 = J00*vm[0*4+0] + J02*vm[0*4+2];
    float Tm01 = J00*vm[1*4+0] + J02*vm[1*4+2];
    float Tm02 = J00*vm[2*4+0] + J02*vm[2*4+2];
    float Tm10 = J11*vm[0*4+1] + J12*vm[0*4+2];
    float Tm11 = J11*vm[1*4+1] + J12*vm[1*4+2];
    float Tm12 = J11*vm[2*4+1] + J12*vm[2*4+2];

    // cov2d = Tm Sigma Tm^T
    float M00 = Tm00*S00 + Tm01*S01 + Tm02*S02;
    float M01 = Tm00*S01 + Tm01*S11 + Tm02*S12;
    float M02 = Tm00*S02 + Tm01*S12 + Tm02*S22;
    float M10 = Tm10*S00 + Tm11*S01 + Tm12*S02;
    float M11 = Tm10*S01 + Tm11*S11 + Tm12*S12;
    float M12 = Tm10*S02 + Tm11*S12 + Tm12*S22;
    float c00 = M00*Tm00 + M01*Tm01 + M02*Tm02 + 0.3f;
    float c01 = M00*Tm10 + M01*Tm11 + M02*Tm12;
    float c11 = M10*Tm10 + M11*Tm11 + M12*Tm12 + 0.3f;

    float det = c00*c11 - c01*c01;
    bool  valid = (tz > 0.2f) && (det > 0.0f);
    float inv_det = (det != 0.0f) ? (1.0f / det) : 0.0f;
    float ca = c11 * inv_det, cb = -c01 * inv_det, cc = c00 * inv_det;

    float mid  = 0.5f * (c00 + c11);
    float lam1 = mid + sqrtf(fmaxf(0.1f, mid*mid - det));
    float radii = valid ? ceilf(3.0f * sqrtf(lam1)) : 0.0f;

    float px = ((ndx + 1.0f) * (float)W_IMG - 1.0f) * 0.5f;
    float py = ((ndy + 1.0f) * (float)H_IMG - 1.0f) * 0.5f;

    // power(n,p) = A . [1, x, y, x^2, xy, y^2, 0, 0]
    float A0 = -0.5f*(ca*px*px + cc*py*py) - cb*px*py;
    float A1 = ca*px + cb*py;
    float A2 = cc*py + cb*px;
    float A3 = -0.5f*ca, A4 = -cb, A5 = -0.5f*cc;

    ws[OFF_KEY + i] = valid ? tz : __builtin_inff();
    float* a = ws + OFF_A + i*8;
    a[0]=A0; a[1]=A1; a[2]=A2; a[3]=A3; a[4]=A4; a[5]=A5; a[6]=0.f; a[7]=0.f;
    ws[OFF_COL + i*3+0] = cols[i*3+0];
    ws[OFF_COL + i*3+1] = cols[i*3+1];
    ws[OFF_COL + i*3+2] = cols[i*3+2];
    ws[OFF_OP + i] = valid ? ops[i] : 0.0f;   // gate: alpha==0 for invalid
    ws[OFF_TZ + i] = tz;

    out[OUT_RADII + i] = radii;
    out[OUT_PVIEW + i*3+0] = pv0;
    out[OUT_PVIEW + i*3+1] = pv1;
    out[OUT_PVIEW + i*3+2] = tz;
}

// ------------------- kernel 2: stable depth rank + scatter -----------------
__global__ void rank_scatter_kernel(float* __restrict__ ws)
{
    __shared__ float keys[N_G];
    int t = threadIdx.x;
    keys[t] = ws[OFF_KEY + t];
    __syncthreads();
    float mk = keys[t];
    int rank = 0;
    for (int j = 0; j < N_G; ++j) {
        float kj = keys[j];
        rank += (kj < mk) || (kj == mk && j < t);   // stable argsort rank
    }
    float* sa = ws + OFF_SA + rank*8;
    const float* a = ws + OFF_A + t*8;
    #pragma unroll
    for (int k = 0; k < 8; ++k) sa[k] = a[k];
    ws[OFF_SCOL + rank*3+0] = ws[OFF_COL + t*3+0];
    ws[OFF_SCOL + rank*3+1] = ws[OFF_COL + t*3+1];
    ws[OFF_SCOL + rank*3+2] = ws[OFF_COL + t*3+2];
    ws[OFF_SOP + rank] = ws[OFF_OP + t];
    ws[OFF_STZ + rank] = ws[OFF_TZ + t];
}

// --------------------------- kernel 3: WMMA blend --------------------------
__global__ void __launch_bounds__(32)
blend_kernel(const float* __restrict__ ws, const float* __restrict__ bg,
             float* __restrict__ out)
{
    const float* sA   = ws + OFF_SA;
    const float* sCol = ws + OFF_SCOL;
    const float* sOp  = ws + OFF_SOP;
    const float* sTz  = ws + OFF_STZ;

    const int lane = threadIdx.x;
    const int half = lane >> 4;        // 0: D rows 0-7, 1: D rows 8-15
    const int col  = lane & 15;        // pixel column within tile / A row M
    const int base = blockIdx.x * 32;  // 32 pixels per wave (two 16-px tiles)

    // B(p) = [1, x, y, x^2, xy, y^2, 0, 0] for both tiles (built once)
    int q1 = base + col, q2 = base + 16 + col;
    float x1 = (float)(q1 & (W_IMG-1)), y1 = (float)(q1 >> 7);
    float x2 = (float)(q2 & (W_IMG-1)), y2 = (float)(q2 >> 7);
    // B 4x16 layout mirrors A: VGPR0 = K0|K2, VGPR1 = K1|K3
    v2f b1k1, b1k2, b2k1, b2k2;
    if (half == 0) {
        b1k1 = (v2f){1.f, x1};        b1k2 = (v2f){x1*y1, y1*y1};
        b2k1 = (v2f){1.f, x2};        b2k2 = (v2f){x2*y2, y2*y2};
    } else {
        b1k1 = (v2f){y1, x1*x1};      b1k2 = (v2f){0.f, 0.f};
        b2k1 = (v2f){y2, x2*x2};      b2k2 = (v2f){0.f, 0.f};
    }

    // lane-addressed A slices: row (gbase+col), K-pairs (2*half) and (4+2*half)
    const float* aBase1 = sA + (size_t)col * 8 + 2*half;       // K0/K1 | K2/K3
    const float* aBase2 = sA + (size_t)col * 8 + 4 + 2*half;   // K4/K5 | K6/K7

    float T = 1.0f, Tfin = 1.0f;
    float accR = 0.f, accG = 0.f, accB = 0.f, accD = 0.f;

    for (int g = 0; g < N_G/16; ++g) {
        int gbase = g * 16;
        size_t rowoff = (size_t)gbase * 8;
        if (g < N_G/16 - 1)
            __builtin_prefetch(aBase1 + rowoff + 128, 0, 0);   // global_prefetch_b8

        // direct 8-byte loads: exactly the two K-pairs this lane contributes
        v2f aK1 = *(const v2f*)(aBase1 + rowoff);
        v2f aK2 = *(const v2f*)(aBase2 + rowoff);

        v8f z = {0.f,0.f,0.f,0.f,0.f,0.f,0.f,0.f};
        // tile 1: 16 gaussians x 16 pixels, K=8 via two chained K=4 WMMAs
        v8f d1 = __builtin_amdgcn_wmma_f32_16x16x4_f32(false, aK1, false, b1k1, (short)0, z,  false, false);
        d1     = __builtin_amdgcn_wmma_f32_16x16x4_f32(false, aK2, false, b1k2, (short)0, d1, false, false);
        // tile 2
        v8f d2 = __builtin_amdgcn_wmma_f32_16x16x4_f32(false, aK1, false, b2k1, (short)0, z,  false, false);
        d2     = __builtin_amdgcn_wmma_f32_16x16x4_f32(false, aK2, false, b2k2, (short)0, d2, false, false);

        // cross-half swap: lane gets all 16 ordered powers for pixel base+lane
        float pw[16];
        #pragma unroll
        for (int r = 0; r < 8; ++r) {
            float own1 = d1[r], own2 = d2[r];
            float send = half ? own1 : own2;
            float recv = __shfl_xor(send, 16, 32);
            pw[r]   = half ? recv : own1;
            pw[8+r] = half ? own2 : recv;
        }

        // ordered transmittance scan; gaussian params are wave-uniform loads
        #pragma unroll
        for (int m = 0; m < 16; ++m) {
            int gi = gbase + m;
            float p = pw[m];
            float e = __expf(p);
            float av = sOp[gi] * e;
            float alpha = (p <= 0.0f && av >= (1.0f/255.0f)) ? fminf(0.99f, av) : 0.0f;
            float Tn = T * (1.0f - alpha);
            bool live = (Tn >= 1e-4f);
            float w = live ? alpha * T : 0.0f;
            Tfin = live ? Tn : Tfin;
            accR += w * sCol[gi*3+0];
            accG += w * sCol[gi*3+1];
            accB += w * sCol[gi*3+2];
            accD += w * sTz[gi];
            T = Tn;
        }
    }

    int p = base + lane;
    out[p]              = accR + Tfin * bg[0];
    out[16384 + p]      = accG + Tfin * bg[1];
    out[32768 + p]      = accB + Tfin * bg[2];
    out[OUT_DEPTH + p]  = accD;
    out[OUT_ALPHA + p]  = 1.0f - Tfin;
}

// ------------------------------- launcher ----------------------------------
extern "C" void kernel_launch(void* const* d_in, const int* in_sizes, int n_in,
                              void* d_out, int out_size, void* d_ws, size_t ws_size,
                              hipStream_t stream)
{
    const float* means  = (const float*)d_in[0];
    const float* cols   = (const float*)d_in[1];
    const float* ops    = (const float*)d_in[2];
    const float* scales = (const float*)d_in[3];
    const float* rots   = (const float*)d_in[4];
    const float* vm     = (const float*)d_in[5];
    const float* pm     = (const float*)d_in[6];
    const float* bg     = (const float*)d_in[7];
    float* out = (float*)d_out;
    float* ws  = (float*)d_ws;

    preprocess_kernel<<<N_G/256, 256, 0, stream>>>(means, cols, ops, scales, rots, vm, pm, ws, out);
    rank_scatter_kernel<<<1, N_G, 0, stream>>>(ws);
    blend_kernel<<<(H_IMG*W_IMG)/32, 32, 0, stream>>>(ws, bg, out);
}